// tree_crf_4277787427616
// MI455X (gfx1250) — compile-verified
//
#include <hip/hip_runtime.h>
#include <math.h>

typedef __attribute__((ext_vector_type(2))) float v2f;
typedef __attribute__((ext_vector_type(8))) float v8f;

constexpr int kM1   = 513;   // original node count
constexpr int kN    = 512;   // minor size (M1 - 1)
constexpr int kLDA  = 520;   // global row stride (padded)
constexpr int kNB   = 32;    // LU block size (8x WMMA K of 4)
constexpr int kPSTR = 34;    // LDS row stride for 32-wide panels (even + conflict-free)

__global__ __launch_bounds__(256)
void tree_logdet_kernel(const float* __restrict__ E_all,
                        const int*   __restrict__ n_nodes,
                        float*       __restrict__ out,
                        float*       __restrict__ ws)
{
    extern __shared__ float smem[];
    float* lds_p   = smem;                                   // kN * kPSTR        (column panel)
    float* lds_bT  = smem + kN * kPSTR;                      // (kN-kNB) * kPSTR  (A12^T, stored negated)
    float* lds_inv = smem + kN * kPSTR + (kN - kNB) * kPSTR; // kNB floats (1/U11 diag)
    float* lds_deg = lds_inv + kNB;                          // kN floats
    float* acc     = lds_deg + kN;                           // 1 float (logdet accumulator)

    const int b    = blockIdx.x;
    const int tid  = threadIdx.x;
    const int lane = tid & 31;
    const int wave = tid >> 5;
    const int n    = n_nodes[b];

    const float* E = E_all + (size_t)b * kM1 * kM1;
    float*       A = ws    + (size_t)b * kN * kLDA;

    // ---- Phase A: degrees deg[r] = sum_{j<n, j!=r} E[r][j], one row per wave ----
    for (int r = 1 + wave; r <= kN; r += 8) {
        float s = 0.f;
        const float* row = E + (size_t)r * kM1;
        for (int j = lane; j < n; j += 32) s += row[j];
        #pragma unroll
        for (int off = 16; off > 0; off >>= 1) s += __shfl_xor(s, off, 32);
        if (lane == 0) {
            if (r < n) s -= row[r];
            lds_deg[r - 1] = s;
        }
    }
    if (tid == 0) *acc = 0.f;
    __syncthreads();

    // ---- Phase B: build Laplacian minor Lm into workspace ----
    for (int idx = tid; idx < kN * kN; idx += 256) {
        int i = idx >> 9, j = idx & (kN - 1);
        int r = i + 1, c = j + 1;
        float v;
        if (i == j) v = (r < n) ? lds_deg[i] : 1.0f;
        else        v = (r < n && c < n) ? -E[(size_t)c * kM1 + r] : 0.0f;
        A[(size_t)i * kLDA + j] = v;
    }
    __threadfence_block();
    __syncthreads();

    // ---- Phase C: blocked LU without pivoting (column-diag-dominant => stable) ----
    for (int k0 = 0; k0 < kN; k0 += kNB) {
        const int rows = kN - k0;

        // C1: stage panel (rows k0..511, cols k0..k0+31) into LDS (coalesced: 32 floats/row)
        for (int idx = tid; idx < rows * kNB; idx += 256) {
            int ii = idx >> 5, kk = idx & 31;
            lds_p[ii * kPSTR + kk] = A[(size_t)(k0 + ii) * kLDA + (k0 + kk)];
        }
        __syncthreads();

        // C2: wave0 factors the 32x32 diagonal block (intra-wave LDS ordering is automatic)
        if (wave == 0) {
            for (int kk = 0; kk < kNB; ++kk) {
                float inv = 1.0f / lds_p[kk * kPSTR + kk];
                int ii = kk + 1 + lane;
                if (ii < kNB) {
                    float lik = lds_p[ii * kPSTR + kk] * inv;
                    lds_p[ii * kPSTR + kk] = lik;
                    for (int jj = kk + 1; jj < kNB; ++jj)
                        lds_p[ii * kPSTR + jj] -= lik * lds_p[kk * kPSTR + jj];
                }
            }
            lds_inv[lane] = 1.0f / lds_p[lane * kPSTR + lane];
        }
        __syncthreads();

        // logdet contribution from this block's pivots (U11 diagonal)
        if (tid == 0) {
            float s = 0.f;
            for (int kk = 0; kk < kNB; kk++) s += logf(lds_p[kk * kPSTR + kk]);
            *acc += s;
        }

        // C3: L21 rows: independent right-solves  x * U11 = a  (register-local, no barriers)
        for (int ii = kNB + tid; ii < rows; ii += 256) {
            float x[kNB];
            float* prow = &lds_p[ii * kPSTR];
            #pragma unroll
            for (int j = 0; j < kNB; ++j) {
                float xv = prow[j];
                #pragma unroll
                for (int m = 0; m < j; ++m) xv -= x[m] * lds_p[m * kPSTR + j];
                x[j] = xv * lds_inv[j];
            }
            #pragma unroll
            for (int j = 0; j < kNB; ++j) prow[j] = x[j];
        }

        const int ncols = kN - k0 - kNB;
        if (ncols > 0) {
            // C4a: stage A12 transposed (division-free, coalesced global reads)
            for (int kk = 0; kk < kNB; ++kk) {
                const float* grow = A + (size_t)(k0 + kk) * kLDA + (k0 + kNB);
                for (int jc = tid; jc < ncols; jc += 256)
                    lds_bT[jc * kPSTR + kk] = grow[jc];
            }
            __syncthreads();   // covers C3 writes + C4a writes

            // C4b: TRSM  U12 = L11^{-1} A12, stored NEGATED (so WMMA accumulates C - L21*U12)
            for (int jc = tid; jc < ncols; jc += 256) {
                float x[kNB];
                #pragma unroll
                for (int kk = 0; kk < kNB; ++kk) {
                    float xv = lds_bT[jc * kPSTR + kk];
                    #pragma unroll
                    for (int m = 0; m < kk; ++m) xv -= lds_p[kk * kPSTR + m] * x[m];
                    x[kk] = xv;
                }
                #pragma unroll
                for (int kk = 0; kk < kNB; ++kk) lds_bT[jc * kPSTR + kk] = -x[kk];
            }
            __syncthreads();

            // C5: trailing update A22 -= L21 * U12 via 8 chained V_WMMA_F32_16X16X4_F32
            const int ntile = ncols >> 4;
            const int hi = lane >> 4;        // lane half selects K-pair / M-offset
            const int lo = lane & 15;        // M index for A, N index for B/C
            const int kp = hi * 2;

            int ti = 0, tj = wave;           // division-free flat tile walk, stride 8
            while (tj >= ntile) { tj -= ntile; ++ti; }
            while (ti < ntile) {
                float* C = A + (size_t)(k0 + kNB + ti * 16) * kLDA + (k0 + kNB + tj * 16);
                v8f c;
                #pragma unroll
                for (int v = 0; v < 8; v++)
                    c[v] = C[(size_t)(v + 8 * hi) * kLDA + lo];

                const float* arow = &lds_p [(kNB + ti * 16 + lo) * kPSTR]; // L21 (lane = M)
                const float* brow = &lds_bT[(tj * 16 + lo) * kPSTR];       // -U12 (lane = N)
                #pragma unroll
                for (int kc = 0; kc < kNB; kc += 4) {
                    v2f a  = *(const v2f*)(arow + kc + kp);
                    v2f bb = *(const v2f*)(brow + kc + kp);
                    c = __builtin_amdgcn_wmma_f32_16x16x4_f32(
                            false, a, false, bb, (short)0, c, false, false);
                }
                #pragma unroll
                for (int v = 0; v < 8; v++)
                    C[(size_t)(v + 8 * hi) * kLDA + lo] = c[v];

                tj += 8;
                while (tj >= ntile) { tj -= ntile; ++ti; }
            }
        }
        __threadfence_block();
        __syncthreads();
    }

    if (tid == 0) out[b] = *acc;
}

extern "C" void kernel_launch(void* const* d_in, const int* in_sizes, int n_in,
                              void* d_out, int out_size, void* d_ws, size_t ws_size,
                              hipStream_t stream) {
    const float* energies = (const float*)d_in[0];
    const int*   n_nodes  = (const int*)d_in[1];
    float*       out      = (float*)d_out;
    float*       ws       = (float*)d_ws;   // needs 256 * 512 * 520 * 4 bytes

    const size_t smem_bytes =
        (size_t)(kN * kPSTR + (kN - kNB) * kPSTR + kNB + kN + 1) * sizeof(float);

    hipLaunchKernelGGL(tree_logdet_kernel, dim3(256), dim3(256), smem_bytes, stream,
                       energies, n_nodes, out, ws);
}